// CrossAttentionBlock_14121852469830
// MI455X (gfx1250) — compile-verified
//
#include <hip/hip_runtime.h>
#include <hip/hip_bf16.h>

// Problem dims (fixed by the reference).
constexpr int Bn = 64;
constexpr int Ln = 1024;
constexpr int Dn = 1024;
constexpr int SPLIT = 8;          // L-splits per batch row for occupancy
constexpr int LC = Ln / SPLIT;    // 128 rows per block
constexpr int NT = 256;           // threads per block (8 wave32)

typedef __attribute__((ext_vector_type(2))) float v2f;
typedef __attribute__((ext_vector_type(8))) float v8f;

// ---------------------------------------------------------------------------
// Phase A: partial masked attention over a 128-row slice of one batch row.
// Produces flash-attention stats (m, s) + unnormalized partial output.
// Bandwidth-bound: streams K and V slices exactly once, float4-coalesced.
// ---------------------------------------------------------------------------
__global__ __launch_bounds__(NT)
void xattn_partial_kernel(const float* __restrict__ q,      // [B, D]
                          const float* __restrict__ K,      // [B, L, D]
                          const float* __restrict__ V,      // [B, L, D]
                          const int*   __restrict__ mask,   // [B, L]
                          float* __restrict__ o_part,       // [SPLIT, B, D]
                          float* __restrict__ ms_part)      // [SPLIT, B, 2]
{
  __shared__ float q_s[Dn];
  __shared__ float p_s[LC];
  __shared__ float red_s[NT / 32];

  const int b    = blockIdx.x;
  const int sp   = blockIdx.y;
  const int tid  = threadIdx.x;
  const int lane = tid & 31;
  const int wave = tid >> 5;
  const float NEG_INF = -__builtin_inff();

  // Stage query row into LDS (256 threads x float4 = 1024 floats).
  const float4* q4 = reinterpret_cast<const float4*>(q + (size_t)b * Dn);
  reinterpret_cast<float4*>(q_s)[tid] = q4[tid];
  __syncthreads();

  const int l0 = sp * LC;

  // Scores: each wave cooperatively computes dot(q, K[b,l]) for 16 rows.
  const float4* qs4 = reinterpret_cast<const float4*>(q_s);
  for (int r = wave; r < LC; r += NT / 32) {
    const int l = l0 + r;
    const float4* k4 =
        reinterpret_cast<const float4*>(K + ((size_t)b * Ln + l) * Dn);
    float acc = 0.0f;
#pragma unroll
    for (int j = 0; j < Dn / 128; ++j) {          // 8 iterations
      float4 kv = k4[lane + 32 * j];              // 512B contiguous per step
      float4 qv = qs4[lane + 32 * j];
      acc += kv.x * qv.x + kv.y * qv.y + kv.z * qv.z + kv.w * qv.w;
    }
#pragma unroll
    for (int off = 16; off > 0; off >>= 1)
      acc += __shfl_xor(acc, off, 32);
    if (lane == 0) {
      const bool valid = mask[(size_t)b * Ln + l] != 0;
      p_s[r] = valid ? acc * 0.03125f : NEG_INF;  // 1/sqrt(1024)
    }
  }
  __syncthreads();

  // Block-wide max over the 128 scores.
  float v = (tid < LC) ? p_s[tid] : NEG_INF;
  float mx = v;
#pragma unroll
  for (int off = 16; off > 0; off >>= 1)
    mx = fmaxf(mx, __shfl_xor(mx, off, 32));
  if (lane == 0) red_s[wave] = mx;
  __syncthreads();
  float m_loc = red_s[0];
#pragma unroll
  for (int i = 1; i < NT / 32; ++i) m_loc = fmaxf(m_loc, red_s[i]);

  // exp(score - m); fully-masked slice (m == -inf) yields all zeros, no NaN.
  const float mclamp = (m_loc == NEG_INF) ? 0.0f : m_loc;
  const float e = (tid < LC) ? __expf(v - mclamp) : 0.0f;
  __syncthreads();                     // red_s / p_s reuse barrier
  if (tid < LC) p_s[tid] = e;
  float sm = e;
#pragma unroll
  for (int off = 16; off > 0; off >>= 1)
    sm += __shfl_xor(sm, off, 32);
  if (lane == 0) red_s[wave] = sm;
  __syncthreads();
  float s_loc = 0.0f;
#pragma unroll
  for (int i = 0; i < NT / 32; ++i) s_loc += red_s[i];

  // Partial output: o[d] = sum_l e[l] * V[b,l,d]; thread owns 4 columns.
  const float4* V4 =
      reinterpret_cast<const float4*>(V + ((size_t)b * Ln + l0) * Dn);
  float4 acc = make_float4(0.f, 0.f, 0.f, 0.f);
  for (int r = 0; r < LC; ++r) {
    const float w  = p_s[r];
    const float4 vv = V4[(size_t)r * (Dn / 4) + tid];
    acc.x += w * vv.x; acc.y += w * vv.y;
    acc.z += w * vv.z; acc.w += w * vv.w;
  }
  float4* O4 =
      reinterpret_cast<float4*>(o_part + ((size_t)sp * Bn + b) * Dn);
  O4[tid] = acc;
  if (tid == 0) {
    ms_part[((size_t)sp * Bn + b) * 2 + 0] = m_loc;
    ms_part[((size_t)sp * Bn + b) * 2 + 1] = s_loc;
  }
}

// ---------------------------------------------------------------------------
// Phase B: merge SPLIT partials with exp(m_i - M) weights -> ctx = softmax @ V
// ---------------------------------------------------------------------------
__global__ __launch_bounds__(NT)
void xattn_combine_kernel(const float* __restrict__ o_part,   // [SPLIT, B, D]
                          const float* __restrict__ ms_part,  // [SPLIT, B, 2]
                          float* __restrict__ ctx)            // [B, D]
{
  const int b   = blockIdx.x;
  const int tid = threadIdx.x;
  const float NEG_INF = -__builtin_inff();

  float m[SPLIT], s[SPLIT], w[SPLIT];
  float M = NEG_INF;
#pragma unroll
  for (int i = 0; i < SPLIT; ++i) {
    m[i] = ms_part[((size_t)i * Bn + b) * 2 + 0];
    s[i] = ms_part[((size_t)i * Bn + b) * 2 + 1];
    M = fmaxf(M, m[i]);
  }
  float S = 0.0f;
#pragma unroll
  for (int i = 0; i < SPLIT; ++i) {
    w[i] = (m[i] == NEG_INF) ? 0.0f : __expf(m[i] - M);
    S += w[i] * s[i];
  }
  const float inv = 1.0f / S;   // mask guarantees >=1 valid token per row

  float4 acc = make_float4(0.f, 0.f, 0.f, 0.f);
#pragma unroll
  for (int i = 0; i < SPLIT; ++i) {
    const float4 ov = reinterpret_cast<const float4*>(
        o_part + ((size_t)i * Bn + b) * Dn)[tid];
    acc.x += w[i] * ov.x; acc.y += w[i] * ov.y;
    acc.z += w[i] * ov.z; acc.w += w[i] * ov.w;
  }
  reinterpret_cast<float4*>(ctx + (size_t)b * Dn)[tid] =
      make_float4(acc.x * inv, acc.y * inv, acc.z * inv, acc.w * inv);
}

// ---------------------------------------------------------------------------
// Phase C: out = q_in + ctx @ W^T + bias  via V_WMMA_F32_16X16X4_F32.
// A = ctx (M=64, K=1024), B[k,n] = W[n,k] (K=1024, N=1024).
// One 16x16 output tile per wave; 256 waves total (32 blocks x 8 waves).
// ISA layouts (32-bit operands, wave32):
//   A 16x4 : lane<16 -> row M=lane,   v0=K0, v1=K1 ; lane>=16 -> K2,K3
//   B 4x16 : lane<16 -> col N=lane,   v0=K0, v1=K1 ; lane>=16 -> K2,K3
//   C/D    : lane<16 -> col N=lane, VGPR r -> row m0+r; lane>=16 -> m0+8+r
// ---------------------------------------------------------------------------
__global__ __launch_bounds__(NT)
void proj_wmma_kernel(const float* __restrict__ ctx,   // [B, D]
                      const float* __restrict__ W,     // [D, D] (row j = out dim)
                      const float* __restrict__ bias,  // [D]
                      const float* __restrict__ q_in,  // [B, D]
                      float* __restrict__ out)         // [B, D]
{
  const int tid  = threadIdx.x;
  const int lane = tid & 31;
  const int wave = tid >> 5;
  const int gw   = blockIdx.x * (NT / 32) + wave;   // 0..255
  const int ntiles = Dn / 16;                       // 64
  const int m0 = (gw / ntiles) * 16;                // 0,16,32,48
  const int n0 = (gw % ntiles) * 16;

  const int row = lane & 15;
  const int kh  = (lane >> 4) << 1;                 // 0 or 2

  const v2f* a2 = reinterpret_cast<const v2f*>(ctx + (size_t)(m0 + row) * Dn + kh);
  const v2f* b2 = reinterpret_cast<const v2f*>(W   + (size_t)(n0 + row) * Dn + kh);

  v8f c = {0.f, 0.f, 0.f, 0.f, 0.f, 0.f, 0.f, 0.f};
  for (int k = 0; k < Dn; k += 4) {
    const v2f a  = a2[k >> 1];
    const v2f bb = b2[k >> 1];
    c = __builtin_amdgcn_wmma_f32_16x16x4_f32(
        /*neg_a=*/false, a, /*neg_b=*/false, bb,
        /*c_mod=*/(short)0, c, /*reuse_a=*/false, /*reuse_b=*/false);
  }

  const int n  = n0 + row;
  const int mb = m0 + ((lane >> 4) << 3);
  const float bv = bias[n];
#pragma unroll
  for (int r = 0; r < 8; ++r) {
    const int m = mb + r;
    out[(size_t)m * Dn + n] = c[r] + q_in[(size_t)m * Dn + n] + bv;
  }
}

// ---------------------------------------------------------------------------
extern "C" void kernel_launch(void* const* d_in, const int* in_sizes, int n_in,
                              void* d_out, int out_size, void* d_ws, size_t ws_size,
                              hipStream_t stream) {
  (void)in_sizes; (void)n_in; (void)out_size; (void)ws_size;

  const float* img_q    = (const float*)d_in[0];
  const float* txt_q    = (const float*)d_in[1];
  const float* K_img    = (const float*)d_in[2];
  const float* V_img    = (const float*)d_in[3];
  const int*   img_mask = (const int*)  d_in[4];
  const float* K_txt    = (const float*)d_in[5];
  const float* V_txt    = (const float*)d_in[6];
  const int*   txt_mask = (const int*)  d_in[7];
  const float* W_img    = (const float*)d_in[8];
  const float* b_img    = (const float*)d_in[9];
  const float* W_txt    = (const float*)d_in[10];
  const float* b_txt    = (const float*)d_in[11];

  float* img_q_out = (float*)d_out;               // [B, D]
  float* txt_q_out = (float*)d_out + (size_t)Bn * Dn;

  // Workspace layout (~2.3 MB, 16B-aligned chunks).
  float* o_part  = (float*)d_ws;                          // SPLIT*B*D
  float* ms_part = o_part + (size_t)SPLIT * Bn * Dn;      // SPLIT*B*2
  float* ctx     = ms_part + (size_t)SPLIT * Bn * 2;      // B*D

  const dim3 gA(Bn, SPLIT);
  const dim3 blk(NT);

  // Phase 1: ctx_img = attn(txt_q, K_img, V_img, img_mask);
  //          img_q_out = img_q + ctx_img @ W_img^T + b_img
  xattn_partial_kernel<<<gA, blk, 0, stream>>>(txt_q, K_img, V_img, img_mask,
                                               o_part, ms_part);
  xattn_combine_kernel<<<Bn, blk, 0, stream>>>(o_part, ms_part, ctx);
  proj_wmma_kernel<<<32, blk, 0, stream>>>(ctx, W_img, b_img, img_q, img_q_out);

  // Phase 2: ctx_txt = attn(img_q_out, K_txt, V_txt, txt_mask);
  //          txt_q_out = txt_q + ctx_txt @ W_txt^T + b_txt
  xattn_partial_kernel<<<gA, blk, 0, stream>>>(img_q_out, K_txt, V_txt, txt_mask,
                                               o_part, ms_part);
  xattn_combine_kernel<<<Bn, blk, 0, stream>>>(o_part, ms_part, ctx);
  proj_wmma_kernel<<<32, blk, 0, stream>>>(ctx, W_txt, b_txt, txt_q, txt_q_out);
}